// RelationshipEncoder_44341242364566
// MI455X (gfx1250) — compile-verified
//
#include <hip/hip_runtime.h>

typedef float v2f __attribute__((ext_vector_type(2)));
typedef float v8f __attribute__((ext_vector_type(8)));

#define D_MODEL   256
#define BATCH_A   4096
#define BATCH_B   8192
#define NKEYS     4
#define NBUCKETS  1024

// ---------------------------------------------------------------------------
// Hash helpers: hash = (sum of 4 int32 keys) % 1024   (keys are non-negative)
// ---------------------------------------------------------------------------
__device__ __forceinline__ int hash_row(const int* __restrict__ keys, int row) {
    int s = 0;
#pragma unroll
    for (int k = 0; k < NKEYS; ++k) s += keys[row * NKEYS + k];
    return (int)(((unsigned)s) & (NBUCKETS - 1));
}

__global__ void hash_b_kernel(const int* __restrict__ keys_b, int* __restrict__ hash_b) {
    int j = blockIdx.x * blockDim.x + threadIdx.x;
    if (j < BATCH_B) hash_b[j] = hash_row(keys_b, j);
}

// ---------------------------------------------------------------------------
// Deterministic per-bucket reduction: block g sums all b-rows with hash == g.
// hash_b[j] is a uniform (scalar) load per iteration -> L2 resident.
// ---------------------------------------------------------------------------
__global__ __launch_bounds__(D_MODEL)
void bucket_sum_kernel(const float* __restrict__ emb_b,
                       const int* __restrict__ hash_b,
                       float* __restrict__ bucket_sums,
                       int* __restrict__ counts) {
    const int g = blockIdx.x;     // bucket id
    const int t = threadIdx.x;    // feature index
    float acc = 0.0f;
    int cnt = 0;
    for (int j = 0; j < BATCH_B; ++j) {
        if (hash_b[j] == g) {
            acc += emb_b[(size_t)j * D_MODEL + t];
            ++cnt;
        }
    }
    bucket_sums[(size_t)g * D_MODEL + t] = acc;
    if (t == 0) counts[g] = cnt;
}

// ---------------------------------------------------------------------------
// combined[i] = [ emb_a[i] , bucket_sum[hash_a[i]] / max(count,1) ]
// ---------------------------------------------------------------------------
__global__ __launch_bounds__(D_MODEL)
void build_combined_kernel(const float* __restrict__ emb_a,
                           const int* __restrict__ keys_a,
                           const float* __restrict__ bucket_sums,
                           const int* __restrict__ counts,
                           float* __restrict__ combined) {
    const int i = blockIdx.x;
    const int t = threadIdx.x;
    const int ha  = hash_row(keys_a, i);
    const int cnt = counts[ha];
    const float scale = 1.0f / (float)(cnt < 1 ? 1 : cnt);
    combined[(size_t)i * (2 * D_MODEL) + t]           = emb_a[(size_t)i * D_MODEL + t];
    combined[(size_t)i * (2 * D_MODEL) + D_MODEL + t] =
        bucket_sums[(size_t)ha * D_MODEL + t] * scale;
}

// ---------------------------------------------------------------------------
// FP32 WMMA GEMM:  C = act(A[MxK] @ B[KxN] + bias)   (row-major everywhere)
// Block tile 64x64, 256 threads = 8 waves (wave32) arranged 4(m) x 2(n);
// each wave computes a 16x32 slab as two 16x16 V_WMMA_F32_16X16X4_F32 tiles.
// Fragment layouts per CDNA5 ISA 7.12.2:
//   A 16x4 : lane = m + 16*(k>>1), vgpr = k&1      -> one b64 LDS read/lane
//   B 4x16 : lane = n + 16*(k>>1), vgpr = k&1
//   C 16x16: vgpr r -> m = r + 8*(lane>>4), n = lane&15
// ---------------------------------------------------------------------------
#define BM 64
#define BN 64
#define KT 16
#define AS_LD 20   // pad: 80B row stride (16B aligned, spreads banks)
#define BS_LD 68   // pad: 272B row stride (16B aligned, 4-bank shift per row)

__global__ __launch_bounds__(256)
void wmma_gemm_bias_kernel(const float* __restrict__ A,
                           const float* __restrict__ B,
                           const float* __restrict__ bias,
                           float* __restrict__ C,
                           int M, int N, int K, int relu) {
    __shared__ __align__(16) float As[BM][AS_LD];
    __shared__ __align__(16) float Bs[KT][BS_LD];

    const int tid  = threadIdx.x;
    const int lane = tid & 31;
    const int wave = tid >> 5;     // 0..7
    const int wm   = wave >> 1;    // 0..3  (16-row slab)
    const int wn   = wave & 1;     // 0..1  (32-col slab)
    const int hi   = lane >> 4;    // K-half / M-half selector
    const int l15  = lane & 15;

    const int bm = blockIdx.y * BM;
    const int bn = blockIdx.x * BN;

    v8f c0 = {};  // cols [wn*32 + 0..15]
    v8f c1 = {};  // cols [wn*32 + 16..31]

    // global->LDS staging indices (one float4 each per thread per K-step)
    const int ar = tid >> 2;          // 0..63
    const int ac = (tid & 3) * 4;     // 0,4,8,12
    const int br = tid >> 4;          // 0..15
    const int bc = (tid & 15) * 4;    // 0..60

    for (int k0 = 0; k0 < K; k0 += KT) {
        const float4 av = *(const float4*)&A[(size_t)(bm + ar) * K + (k0 + ac)];
        const float4 bv = *(const float4*)&B[(size_t)(k0 + br) * N + (bn + bc)];
        __syncthreads();                       // previous tile fully consumed
        *(float4*)&As[ar][ac] = av;
        *(float4*)&Bs[br][bc] = bv;
        __syncthreads();

#pragma unroll
        for (int kk = 0; kk < KT; kk += 4) {
            const int kb = kk + 2 * hi;
            v2f a = *(const v2f*)&As[wm * 16 + l15][kb];   // ds_load_b64
            v2f b0, b1;
            b0.x = Bs[kb    ][wn * 32 + l15];
            b0.y = Bs[kb + 1][wn * 32 + l15];
            b1.x = Bs[kb    ][wn * 32 + 16 + l15];
            b1.y = Bs[kb + 1][wn * 32 + 16 + l15];
            c0 = __builtin_amdgcn_wmma_f32_16x16x4_f32(false, a, false, b0,
                                                       (short)0, c0, false, false);
            c1 = __builtin_amdgcn_wmma_f32_16x16x4_f32(false, a, false, b1,
                                                       (short)0, c1, false, false);
        }
    }

    // epilogue: bias (+ optional ReLU), direct global store
    const int n0 = bn + wn * 32 + l15;
    const float bia0 = bias[n0];
    const float bia1 = bias[n0 + 16];
#pragma unroll
    for (int r = 0; r < 8; ++r) {
        const int m = bm + wm * 16 + r + 8 * hi;
        float v0 = c0[r] + bia0;
        float v1 = c1[r] + bia1;
        if (relu) { v0 = fmaxf(v0, 0.0f); v1 = fmaxf(v1, 0.0f); }
        C[(size_t)m * N + n0]      = v0;
        C[(size_t)m * N + n0 + 16] = v1;
    }
}

// ---------------------------------------------------------------------------
// Launch pipeline
// ---------------------------------------------------------------------------
extern "C" void kernel_launch(void* const* d_in, const int* in_sizes, int n_in,
                              void* d_out, int out_size, void* d_ws, size_t ws_size,
                              hipStream_t stream) {
    const float* emb_a  = (const float*)d_in[0];
    const float* emb_b  = (const float*)d_in[1];
    const int*   keys_a = (const int*)d_in[2];
    const int*   keys_b = (const int*)d_in[3];
    const float* W1 = (const float*)d_in[4];
    const float* b1 = (const float*)d_in[5];
    const float* W2 = (const float*)d_in[6];
    const float* b2 = (const float*)d_in[7];
    const float* Wo = (const float*)d_in[8];
    const float* bo = (const float*)d_in[9];
    float* out = (float*)d_out;

    // workspace layout (floats)
    float* ws          = (float*)d_ws;
    int*   hash_b      = (int*)ws;                                  // 8192
    float* bucket_sums = ws + BATCH_B;                              // 1024*256
    int*   counts      = (int*)(bucket_sums + NBUCKETS * D_MODEL);  // 1024
    float* combined    = bucket_sums + NBUCKETS * D_MODEL + NBUCKETS; // 4096*512
    float* h           = combined + (size_t)BATCH_A * 2 * D_MODEL;    // 4096*512
    float* rel         = h        + (size_t)BATCH_A * 2 * D_MODEL;    // 4096*256

    hash_b_kernel<<<(BATCH_B + 255) / 256, 256, 0, stream>>>(keys_b, hash_b);
    bucket_sum_kernel<<<NBUCKETS, D_MODEL, 0, stream>>>(emb_b, hash_b, bucket_sums, counts);
    build_combined_kernel<<<BATCH_A, D_MODEL, 0, stream>>>(emb_a, keys_a, bucket_sums,
                                                           counts, combined);

    // h   = relu(combined @ W1 + b1)   : 4096x512x512
    wmma_gemm_bias_kernel<<<dim3(512 / BN, BATCH_A / BM), 256, 0, stream>>>(
        combined, W1, b1, h, BATCH_A, 512, 512, 1);
    // rel = h @ W2 + b2                : 4096x512->256
    wmma_gemm_bias_kernel<<<dim3(256 / BN, BATCH_A / BM), 256, 0, stream>>>(
        h, W2, b2, rel, BATCH_A, 256, 512, 0);
    // out = rel @ Wo + bo              : 4096x256x256
    wmma_gemm_bias_kernel<<<dim3(256 / BN, BATCH_A / BM), 256, 0, stream>>>(
        rel, Wo, bo, out, BATCH_A, 256, 256, 0);
}